// Attention_12463995093474
// MI455X (gfx1250) — compile-verified
//
#include <hip/hip_runtime.h>
#include <hip/hip_bf16.h>
#include <math.h>

#define Bc 16
#define Tc 200
#define Dc 128
#define Hc 4
#define H_ABSc 2
#define H_RELc 2
#define DKc 32

typedef __attribute__((ext_vector_type(2))) float v2f;
typedef __attribute__((ext_vector_type(4))) float v4f;
typedef __attribute__((ext_vector_type(8))) float v8f;

// One K=4 step of the fp32 WMMA: D(16x16) += A(16x4) * B(4x16)
// A frag: a.x = A[lane%16][k+2*(lane/16)], a.y = A[lane%16][k+2*(lane/16)+1]
// B frag: b.x = B[k+2*(lane/16)][lane%16], b.y = B[k+2*(lane/16)+1][lane%16]
__device__ __forceinline__ v8f wmma_step(v8f acc, float a0, float a1, float b0, float b1) {
    v2f a = {a0, a1};
    v2f b = {b0, b1};
    return __builtin_amdgcn_wmma_f32_16x16x4_f32(false, a, false, b, (short)0, acc,
                                                 false, false);
}

// ---------------------------------------------------------------------------
// Kernel 1: X(3200x128) @ W(128x128) + bias -> head-split out[B][H][T][DK]
// grid.x = 200 row tiles, block = 256 (8 waves x 16 cols = 128 cols)
// ---------------------------------------------------------------------------
__global__ void proj_split_heads(const float* __restrict__ X, const float* __restrict__ W,
                                 const float* __restrict__ bias, float* __restrict__ out) {
    int lane = threadIdx.x & 31, wave = threadIdx.x >> 5;
    int half = lane >> 4, lr = lane & 15;
    int m0 = blockIdx.x * 16;
    int n0 = wave * 16;
    const float* arow = X + (size_t)(m0 + lr) * Dc;
    v8f acc = {};
    for (int k = 0; k < Dc; k += 4) {
        int kk = k + 2 * half;
        acc = wmma_step(acc, arow[kk], arow[kk + 1],
                        W[(size_t)kk * Dc + n0 + lr],
                        W[(size_t)(kk + 1) * Dc + n0 + lr]);
    }
    int n = n0 + lr;
    int h = n >> 5, dk = n & 31;
    float bb = bias[n];
#pragma unroll
    for (int v = 0; v < 8; ++v) {
        int m = m0 + v + 8 * half;
        int b = m / Tc, t = m % Tc;
        out[(((size_t)b * Hc + h) * Tc + t) * DKc + dk] = acc[v] + bb;
    }
}

// ---------------------------------------------------------------------------
// Kernel 2: Aq = X_head + abs_kernel[h] @ Wt[h] + bias[h]
// grid = (200, H_ABS), block = 64 (2 waves, N=32)
// ---------------------------------------------------------------------------
__global__ void abs_aug(const float* __restrict__ AK, const float* __restrict__ Wt,
                        const float* __restrict__ bs, const float* __restrict__ headbuf,
                        float* __restrict__ out) {
    int lane = threadIdx.x & 31, wave = threadIdx.x >> 5;
    int half = lane >> 4, lr = lane & 15;
    int m0 = blockIdx.x * 16;
    int h = blockIdx.y;
    int n0 = wave * 16;
    const float* arow = AK + ((size_t)h * Bc * Tc + m0 + lr) * Dc;
    const float* wbase = Wt + (size_t)h * Dc * DKc;
    v8f acc = {};
    for (int k = 0; k < Dc; k += 4) {
        int kk = k + 2 * half;
        acc = wmma_step(acc, arow[kk], arow[kk + 1],
                        wbase[(size_t)kk * DKc + n0 + lr],
                        wbase[(size_t)(kk + 1) * DKc + n0 + lr]);
    }
    int n = n0 + lr;
    float bb = bs[h * DKc + n];
#pragma unroll
    for (int v = 0; v < 8; ++v) {
        int m = m0 + v + 8 * half;
        int b = m / Tc, t = m % Tc;
        size_t xi = (((size_t)b * Hc + h) * Tc + t) * DKc + n;       // heads 0..1 of qkv buf
        size_t oi = (((size_t)b * H_ABSc + h) * Tc + t) * DKc + n;
        out[oi] = acc[v] + bb + headbuf[xi];
    }
}

// ---------------------------------------------------------------------------
// Kernel 3: w'[h,b,i,d] = sum_o (q_rel[b,h,i,o] + rel_bias[h,o]) * rel_k_w[h,d,o]
// grid = (200, H_REL), block = 256 (8 waves, N=128), K=32
// ---------------------------------------------------------------------------
__global__ void rel_w(const float* __restrict__ qbuf, const float* __restrict__ relw,
                      const float* __restrict__ rbias, float* __restrict__ wpr) {
    int lane = threadIdx.x & 31, wave = threadIdx.x >> 5;
    int half = lane >> 4, lr = lane & 15;
    int m0 = blockIdx.x * 16;
    int h = blockIdx.y;
    int n0 = wave * 16;
    int m = m0 + lr;
    int b = m / Tc, i = m % Tc;
    const float* arow = qbuf + (((size_t)b * Hc + (H_ABSc + h)) * Tc + i) * DKc;
    const float* bias = rbias + h * DKc;
    const float* brow = relw + ((size_t)h * Dc + (n0 + lr)) * DKc;  // B[o][n] = rel_k_w[h][n][o]
    v8f acc = {};
    for (int k = 0; k < DKc; k += 4) {
        int kk = k + 2 * half;
        acc = wmma_step(acc, arow[kk] + bias[kk], arow[kk + 1] + bias[kk + 1],
                        brow[kk], brow[kk + 1]);
    }
#pragma unroll
    for (int v = 0; v < 8; ++v) {
        int mm = m0 + v + 8 * half;
        int bb = mm / Tc, ii = mm % Tc;
        wpr[(((size_t)h * Bc + bb) * Tc + ii) * Dc + n0 + lr] = acc[v];
    }
}

// ---------------------------------------------------------------------------
// Kernel 4 (bandwidth-dominant, ~655 MB stream):
// extra[h,b,i,j] = sum_d rel_kernel[h,b,i,j,d] * w'[h,b,i,d] + c[h,b,i]
// grid = H_REL*B*T = 6400 blocks, block = 256
// ---------------------------------------------------------------------------
__global__ void rel_scores(const float* __restrict__ relk, const float* __restrict__ wpr,
                           const float* __restrict__ qbuf, const float* __restrict__ rbias,
                           const float* __restrict__ relb, float* __restrict__ rex) {
    __shared__ float wrow[Dc];
    __shared__ float cval;
    int idx = blockIdx.x;
    int h = idx / (Bc * Tc);
    int rem = idx % (Bc * Tc);
    int b = rem / Tc, i = rem % Tc;
    int t = threadIdx.x;
    if (t < Dc) wrow[t] = wpr[(((size_t)h * Bc + b) * Tc + i) * Dc + t];
    if (t == 0) {
        const float* qrow = qbuf + (((size_t)b * Hc + (H_ABSc + h)) * Tc + i) * DKc;
        float c = 0.f;
        for (int o = 0; o < DKc; ++o)
            c += relb[h * DKc + o] * (qrow[o] + rbias[h * DKc + o]);
        cval = c;
    }
    __syncthreads();
    int j = t;
    if (j < Tc) {
        const v4f* kr = (const v4f*)(relk + ((((size_t)h * Bc + b) * Tc + i) * Tc + j) * Dc);
        float s = 0.f;
#pragma unroll 8
        for (int d4 = 0; d4 < Dc / 4; ++d4) {
            v4f kv = __builtin_nontemporal_load(kr + d4);  // stream, don't pollute L2
            s += kv.x * wrow[4 * d4] + kv.y * wrow[4 * d4 + 1] +
                 kv.z * wrow[4 * d4 + 2] + kv.w * wrow[4 * d4 + 3];
        }
        rex[(((size_t)h * Bc + b) * Tc + i) * Tc + j] = s + cval;
    }
}

// ---------------------------------------------------------------------------
// Kernel 5: attention tail. grid = (13 i-tiles, H, B), block = 128 (4 waves).
// WMMA fp32 QK^T -> LDS, + rel extra, scale, mask, softmax, P.V -> xctx (B,T,D)
// ---------------------------------------------------------------------------
__global__ void attn(const float* __restrict__ qbuf, const float* __restrict__ kbuf,
                     const float* __restrict__ vbuf, const float* __restrict__ Aq,
                     const float* __restrict__ Ak, const float* __restrict__ rex,
                     const int* __restrict__ mask, float* __restrict__ xctx) {
    __shared__ float S[16][208];
    int it = blockIdx.x, h = blockIdx.y, b = blockIdx.z;
    int i0 = it * 16;
    int lane = threadIdx.x & 31, wave = threadIdx.x >> 5;
    int half = lane >> 4, lr = lane & 15;
    const float *Qp, *Kp;
    if (h < H_ABSc) {
        Qp = Aq + ((size_t)b * H_ABSc + h) * Tc * DKc;
        Kp = Ak + ((size_t)b * H_ABSc + h) * Tc * DKc;
    } else {
        Qp = qbuf + ((size_t)b * Hc + h) * Tc * DKc;
        Kp = kbuf + ((size_t)b * Hc + h) * Tc * DKc;
    }
    const float scale = 0.17677669529663687f;  // 1/sqrt(32)

    for (int jt = wave; jt < 13; jt += 4) {
        int j0 = jt * 16;
        int qi = i0 + lr;  bool qv = qi < Tc;
        int kj = j0 + lr;  bool kv = kj < Tc;
        v8f acc = {};
        for (int k = 0; k < DKc; k += 4) {
            int kk = k + 2 * half;
            float a0 = qv ? Qp[(size_t)qi * DKc + kk] : 0.f;
            float a1 = qv ? Qp[(size_t)qi * DKc + kk + 1] : 0.f;
            float b0 = kv ? Kp[(size_t)kj * DKc + kk] : 0.f;
            float b1 = kv ? Kp[(size_t)kj * DKc + kk + 1] : 0.f;
            acc = wmma_step(acc, a0, a1, b0, b1);
        }
#pragma unroll
        for (int v = 0; v < 8; ++v) {
            int r = v + 8 * half;
            int i = i0 + r;
            int j = j0 + lr;
            float s;
            if (i < Tc && j < Tc) {
                s = acc[v];
                if (h >= H_ABSc)
                    s += rex[(((size_t)(h - H_ABSc) * Bc + b) * Tc + i) * Tc + j];
                s *= scale;
                if (mask[((size_t)b * Tc + i) * Tc + j] == 0) s = -1.0e9f;
            } else {
                s = (i < Tc) ? -INFINITY : 0.f;   // -inf pad cols -> exp = 0
            }
            S[r][j] = s;
        }
    }
    __syncthreads();

    // softmax: wave w handles rows 4w..4w+3, lanes stride the 208 columns
    for (int rr = 0; rr < 4; ++rr) {
        int r = wave * 4 + rr;
        float m = -INFINITY;
        for (int c = lane; c < 208; c += 32) m = fmaxf(m, S[r][c]);
        for (int off = 16; off >= 1; off >>= 1) m = fmaxf(m, __shfl_xor(m, off, 32));
        float sum = 0.f;
        for (int c = lane; c < 208; c += 32) {
            float e = __expf(S[r][c] - m);
            S[r][c] = e;
            sum += e;
        }
        for (int off = 16; off >= 1; off >>= 1) sum += __shfl_xor(sum, off, 32);
        float inv = 1.0f / sum;
        for (int c = lane; c < 208; c += 32) S[r][c] *= inv;
    }
    __syncthreads();

    // P.V : out tile 16x32; lanes of a wave share r, span d -> coalesced v loads
    const float* Vp = vbuf + ((size_t)b * Hc + h) * Tc * DKc;
    for (int idx = threadIdx.x; idx < 16 * DKc; idx += 128) {
        int r = idx >> 5, d = idx & 31;
        int i = i0 + r;
        if (i < Tc) {
            float s = 0.f;
            for (int j = 0; j < Tc; ++j) s += S[r][j] * Vp[(size_t)j * DKc + d];
            xctx[((size_t)b * Tc + i) * Dc + h * DKc + d] = s;
        }
    }
}

// ---------------------------------------------------------------------------
// Kernel 6: out = X(3200x128) @ Wo + bo
// ---------------------------------------------------------------------------
__global__ void out_proj(const float* __restrict__ X, const float* __restrict__ W,
                         const float* __restrict__ bias, float* __restrict__ out) {
    int lane = threadIdx.x & 31, wave = threadIdx.x >> 5;
    int half = lane >> 4, lr = lane & 15;
    int m0 = blockIdx.x * 16;
    int n0 = wave * 16;
    const float* arow = X + (size_t)(m0 + lr) * Dc;
    v8f acc = {};
    for (int k = 0; k < Dc; k += 4) {
        int kk = k + 2 * half;
        acc = wmma_step(acc, arow[kk], arow[kk + 1],
                        W[(size_t)kk * Dc + n0 + lr],
                        W[(size_t)(kk + 1) * Dc + n0 + lr]);
    }
    float bb = bias[n0 + lr];
#pragma unroll
    for (int v = 0; v < 8; ++v) {
        int m = m0 + v + 8 * half;
        out[(size_t)m * Dc + n0 + lr] = acc[v] + bb;
    }
}

extern "C" void kernel_launch(void* const* d_in, const int* in_sizes, int n_in,
                              void* d_out, int out_size, void* d_ws, size_t ws_size,
                              hipStream_t stream) {
    const float* query = (const float*)d_in[0];
    const float* key   = (const float*)d_in[1];
    const float* value = (const float*)d_in[2];
    const int*   mask  = (const int*)d_in[3];
    const float* relk  = (const float*)d_in[4];
    const float* absk  = (const float*)d_in[5];
    // d_in[6] = layer (unused)
    const float* Wq = (const float*)d_in[7];
    const float* bq = (const float*)d_in[8];
    const float* Wk = (const float*)d_in[9];
    const float* bk = (const float*)d_in[10];
    const float* Wv = (const float*)d_in[11];
    const float* bv = (const float*)d_in[12];
    const float* aqw = (const float*)d_in[13];
    const float* aqb = (const float*)d_in[14];
    const float* akw = (const float*)d_in[15];
    const float* akb = (const float*)d_in[16];
    const float* rkw = (const float*)d_in[17];
    const float* rkb = (const float*)d_in[18];
    const float* rbias = (const float*)d_in[19];
    const float* Wo = (const float*)d_in[20];
    const float* bo = (const float*)d_in[21];

    float* ws  = (float*)d_ws;
    float* qb_ = ws;                   // B*H*T*DK   = 409600
    float* kb_ = qb_ + 409600;
    float* vb_ = kb_ + 409600;
    float* Aq  = vb_ + 409600;         // B*H_ABS*T*DK = 204800
    float* Ak  = Aq + 204800;
    float* wpr = Ak + 204800;          // H_REL*B*T*D  = 819200
    float* rex = wpr + 819200;         // H_REL*B*T*T  = 1280000 (2560000 reserved)
    float* xct = rex + 2560000;        // B*T*D        = 409600

    proj_split_heads<<<200, 256, 0, stream>>>(query, Wq, bq, qb_);
    proj_split_heads<<<200, 256, 0, stream>>>(key,   Wk, bk, kb_);
    proj_split_heads<<<200, 256, 0, stream>>>(value, Wv, bv, vb_);
    abs_aug<<<dim3(200, H_ABSc), 64, 0, stream>>>(absk, aqw, aqb, qb_, Aq);
    abs_aug<<<dim3(200, H_ABSc), 64, 0, stream>>>(absk, akw, akb, kb_, Ak);
    rel_w<<<dim3(200, H_RELc), 256, 0, stream>>>(qb_, rkw, rbias, wpr);
    rel_scores<<<H_RELc * Bc * Tc, 256, 0, stream>>>(relk, wpr, qb_, rbias, rkb, rex);
    attn<<<dim3(13, Hc, Bc), 128, 0, stream>>>(qb_, kb_, vb_, Aq, Ak, rex, mask, xct);
    out_proj<<<200, 256, 0, stream>>>(xct, Wo, bo, (float*)d_out);
}